// GnnDQNAgent_42700564856884
// MI455X (gfx1250) — compile-verified
//
#include <hip/hip_runtime.h>
#include <hip/hip_bf16.h>

typedef float v2f __attribute__((ext_vector_type(2)));
typedef float v8f __attribute__((ext_vector_type(8)));

// ---------------------------------------------------------------- utilities
__global__ void fill_kernel(float* __restrict__ p, float v, long long count) {
    long long t = (long long)blockIdx.x * blockDim.x + threadIdx.x;
    if (t < count) p[t] = v;
}

// count4 = number of float4 elements
__global__ void fill4_kernel(float4* __restrict__ p, float v, long long count4) {
    long long t = (long long)blockIdx.x * blockDim.x + threadIdx.x;
    if (t < count4) p[t] = make_float4(v, v, v, v);
}

__global__ void deg_accum_kernel(const long long* __restrict__ ei,
                                 const float* __restrict__ w,
                                 float* __restrict__ deg, int E) {
    int e = blockIdx.x * blockDim.x + threadIdx.x;
    if (e < E) {
        int d = (int)ei[(long long)E + e];
        atomicAdd(&deg[d], w[e]);
    }
}

__global__ void rsqrt_kernel(float* __restrict__ d, int n) {
    int i = blockIdx.x * blockDim.x + threadIdx.x;
    if (i < n) {
        float v = d[i];
        d[i] = (v > 0.0f) ? rsqrtf(v) : 0.0f;
    }
}

__global__ void norm_kernel(const long long* __restrict__ ei,
                            const float* __restrict__ w,
                            const float* __restrict__ dinv,
                            float* __restrict__ nrm, int E) {
    int e = blockIdx.x * blockDim.x + threadIdx.x;
    if (e < E) {
        int s = (int)ei[e];
        int d = (int)ei[(long long)E + e];
        nrm[e] = dinv[s] * w[e] * dinv[d];
    }
}

// ------------------------------------------------- WMMA GEMM: M = A @ W
// A: [n,128] f32, W: [128,Cout] f32, M: [n,Cout].  One wave per 16x16 tile.
// blockDim.x = 32 * (Cout/16); gridDim.x = ceil(n/16).
__global__ void gemm128_wmma(const float* __restrict__ A,
                             const float* __restrict__ W,
                             float* __restrict__ M, int n, int Cout) {
    __shared__ float As[16 * 132];              // pad 128 -> 132 (bank spread)
    const int row0 = blockIdx.x << 4;
    for (int idx = threadIdx.x; idx < 16 * 128; idx += blockDim.x) {
        int r = idx >> 7, c = idx & 127;
        int rr = row0 + r; if (rr >= n) rr = n - 1;      // clamp tail loads
        As[r * 132 + c] = A[(long long)rr * 128 + c];
    }
    __syncthreads();

    const int lane    = threadIdx.x & 31;
    const int colbase = (threadIdx.x >> 5) << 4;
    const int mrow    = lane & 15;                // M (for A) / N (for B)
    const int khalf   = (lane >> 4) << 1;         // 0 or 2
    const int col     = colbase + mrow;

    v8f acc = {0.f, 0.f, 0.f, 0.f, 0.f, 0.f, 0.f, 0.f};
    for (int k0 = 0; k0 < 128; k0 += 4) {
        const int kk = k0 + khalf;
        v2f a = *(const v2f*)&As[mrow * 132 + kk];
        v2f b;
        b.x = W[kk * Cout + col];
        b.y = W[(kk + 1) * Cout + col];
        acc = __builtin_amdgcn_wmma_f32_16x16x4_f32(
            false, a, false, b, (short)0, acc, false, false);
    }

    const int rbase = row0 + ((lane >> 4) << 3);
    if (row0 + 16 <= n) {
        // fast path: whole tile in range, unguarded stores
        float* out = M + (long long)rbase * Cout + col;
#pragma unroll
        for (int r = 0; r < 8; ++r) out[(long long)r * Cout] = acc[r];
    } else {
#pragma unroll
        for (int r = 0; r < 8; ++r) {
            int row = rbase + r;
            if (row < n) M[(long long)row * Cout + col] = acc[r];
        }
    }
}

// ----------------------------------------------------------- edge scatter
// agg[dst, c] += norm[e] * m[src, c];  one thread per (edge, 4-col quad).
// qshift = log2(Cout/4): quads per edge. For Cout=128, one wave == one edge.
__global__ void scatter_kernel(const float* __restrict__ m,
                               const long long* __restrict__ ei,
                               const float* __restrict__ nrm,
                               float* __restrict__ agg, int E, int qshift) {
    long long t = (long long)blockIdx.x * blockDim.x + threadIdx.x;
    long long total = (long long)E << qshift;
    if (t >= total) return;
    int e  = (int)(t >> qshift);
    int c4 = (int)(t & ((1 << qshift) - 1));
    int s  = (int)ei[e];
    int d  = (int)ei[(long long)E + e];
    float nv = nrm[e];
    const int cshift = qshift + 2;               // log2(Cout)
    const float4 mv = *(const float4*)&m[((long long)s << cshift) + (c4 << 2)];
    float* ap = &agg[((long long)d << cshift) + (c4 << 2)];
    atomicAdd(ap + 0, nv * mv.x);
    atomicAdd(ap + 1, nv * mv.y);
    atomicAdd(ap + 2, nv * mv.z);
    atomicAdd(ap + 3, nv * mv.w);
}

// out = relu(agg + dinv^2 * m + bias)   (self-loop term + bias + relu)
// one float4 per thread; qshift = log2(Cout/4)
__global__ void finalize_kernel(const float4* __restrict__ agg,
                                const float4* __restrict__ m,
                                const float* __restrict__ dinv,
                                const float4* __restrict__ bias,
                                float4* __restrict__ out, int n, int qshift) {
    long long t = (long long)blockIdx.x * blockDim.x + threadIdx.x;
    long long total = (long long)n << qshift;
    if (t >= total) return;
    int i  = (int)(t >> qshift);
    int c4 = (int)(t & ((1 << qshift) - 1));
    float di = dinv[i];
    float d2 = di * di;
    float4 a = agg[t], mm = m[t], b = bias[c4];
    float4 r;
    r.x = fmaxf(fmaf(d2, mm.x, a.x) + b.x, 0.0f);
    r.y = fmaxf(fmaf(d2, mm.y, a.y) + b.y, 0.0f);
    r.z = fmaxf(fmaf(d2, mm.z, a.z) + b.z, 0.0f);
    r.w = fmaxf(fmaf(d2, mm.w, a.w) + b.w, 0.0f);
    out[t] = r;
}

// ----------------------------------------------------------- dueling heads
// One wave per node: 32 lanes = 32 hidden units of both small MLPs.
__global__ void heads_kernel(const float* __restrict__ h3,
                             const float* __restrict__ Wv1, const float* __restrict__ bv1,
                             const float* __restrict__ Wv2, const float* __restrict__ bv2,
                             const float* __restrict__ Wa1, const float* __restrict__ ba1,
                             const float* __restrict__ Wa2, const float* __restrict__ ba2,
                             float* __restrict__ val, float* __restrict__ adv,
                             float* __restrict__ advsum, int n) {
    __shared__ float sadv[8];
    const int wave = threadIdx.x >> 5;
    const int lane = threadIdx.x & 31;
    const int i = blockIdx.x * 8 + wave;
    float a_contrib = 0.0f;
    if (i < n) {
        float accv = bv1[lane];
        float acca = ba1[lane];
        const float* hrow = h3 + (long long)i * 64;
#pragma unroll
        for (int k = 0; k < 64; ++k) {
            float hk = hrow[k];
            accv = fmaf(hk, Wv1[k * 32 + lane], accv);
            acca = fmaf(hk, Wa1[k * 32 + lane], acca);
        }
        float rv = fmaxf(accv, 0.0f) * Wv2[lane];
        float ra = fmaxf(acca, 0.0f) * Wa2[lane];
#pragma unroll
        for (int off = 16; off > 0; off >>= 1) {
            rv += __shfl_xor(rv, off, 32);
            ra += __shfl_xor(ra, off, 32);
        }
        if (lane == 0) {
            val[i] = rv + bv2[0];
            float av = ra + ba2[0];
            adv[i] = av;
            a_contrib = av;
        }
    }
    if (lane == 0) sadv[wave] = a_contrib;
    __syncthreads();
    if (threadIdx.x == 0) {
        float s = 0.0f;
#pragma unroll
        for (int w = 0; w < 8; ++w) s += sadv[w];
        atomicAdd(advsum, s);
    }
}

__global__ void q_kernel(const float* __restrict__ val, const float* __restrict__ adv,
                         const float* __restrict__ advsum, float* __restrict__ q, int n) {
    int i = blockIdx.x * blockDim.x + threadIdx.x;
    if (i < n) q[i] = val[i] + adv[i] - advsum[0] / (float)n;
}

// ---------------------------------------------------------------- launcher
extern "C" void kernel_launch(void* const* d_in, const int* in_sizes, int n_in,
                              void* d_out, int out_size, void* d_ws, size_t ws_size,
                              hipStream_t stream) {
    const float*     x   = (const float*)d_in[0];
    const long long* ei  = (const long long*)d_in[1];   // int64 [2,E]
    const float*     w   = (const float*)d_in[2];
    const float* W1 = (const float*)d_in[3];  const float* b1 = (const float*)d_in[4];
    const float* W2 = (const float*)d_in[5];  const float* b2 = (const float*)d_in[6];
    const float* W3 = (const float*)d_in[7];  const float* b3 = (const float*)d_in[8];
    const float* Wv1 = (const float*)d_in[9];  const float* bv1 = (const float*)d_in[10];
    const float* Wv2 = (const float*)d_in[11]; const float* bv2 = (const float*)d_in[12];
    const float* Wa1 = (const float*)d_in[13]; const float* ba1 = (const float*)d_in[14];
    const float* Wa2 = (const float*)d_in[15]; const float* ba2 = (const float*)d_in[16];

    const int n = in_sizes[0] / 128;
    const int E = in_sizes[2];

    float* ws = (float*)d_ws;
    long long o = 0;
    float* bufA = ws + o;  o += (long long)n * 128;   // layer activations
    float* bufB = ws + o;  o += (long long)n * 128;   // m = h @ W
    float* bufC = ws + o;  o += (long long)n * 128;   // agg
    float* dinv = ws + o;  o += n;                    // deg -> rsqrt(deg)
    float* nrm  = ws + o;  o += E;                    // per-edge norm
    float* val  = ws + o;  o += n;
    float* adv  = ws + o;  o += n;
    float* asum = ws + o;  o += 1;

    const int T = 256;
    auto blk = [](long long c, int t) { return (unsigned)((c + t - 1) / t); };

    // ---- gcn_norm (shared by all layers)
    fill_kernel<<<blk(n, T), T, 0, stream>>>(dinv, 1.0f, n);          // self-loop w=1
    deg_accum_kernel<<<blk(E, T), T, 0, stream>>>(ei, w, dinv, E);
    rsqrt_kernel<<<blk(n, T), T, 0, stream>>>(dinv, n);
    norm_kernel<<<blk(E, T), T, 0, stream>>>(ei, w, dinv, nrm, E);

    const unsigned gGemm = blk(n, 16);
    const long long n128q = (long long)n * 32;   // n*128/4 quads
    const long long n64q  = (long long)n * 16;   // n*64/4  quads

    // ---- layer 1: x[ N,128 ] -> bufA[ N,128 ]
    gemm128_wmma<<<gGemm, 256, 0, stream>>>(x, W1, bufB, n, 128);
    fill4_kernel<<<blk(n128q, T), T, 0, stream>>>((float4*)bufC, 0.0f, n128q);
    scatter_kernel<<<blk((long long)E * 32, T), T, 0, stream>>>(bufB, ei, nrm, bufC, E, 5);
    finalize_kernel<<<blk(n128q, T), T, 0, stream>>>((const float4*)bufC, (const float4*)bufB,
                                                     dinv, (const float4*)b1, (float4*)bufA, n, 5);

    // ---- layer 2: bufA -> bufA
    gemm128_wmma<<<gGemm, 256, 0, stream>>>(bufA, W2, bufB, n, 128);
    fill4_kernel<<<blk(n128q, T), T, 0, stream>>>((float4*)bufC, 0.0f, n128q);
    scatter_kernel<<<blk((long long)E * 32, T), T, 0, stream>>>(bufB, ei, nrm, bufC, E, 5);
    finalize_kernel<<<blk(n128q, T), T, 0, stream>>>((const float4*)bufC, (const float4*)bufB,
                                                     dinv, (const float4*)b2, (float4*)bufA, n, 5);

    // ---- layer 3: bufA[ N,128 ] -> bufA[ N,64 ]
    gemm128_wmma<<<gGemm, 128, 0, stream>>>(bufA, W3, bufB, n, 64);
    fill4_kernel<<<blk(n64q, T), T, 0, stream>>>((float4*)bufC, 0.0f, n64q);
    scatter_kernel<<<blk((long long)E * 16, T), T, 0, stream>>>(bufB, ei, nrm, bufC, E, 4);
    finalize_kernel<<<blk(n64q, T), T, 0, stream>>>((const float4*)bufC, (const float4*)bufB,
                                                    dinv, (const float4*)b3, (float4*)bufA, n, 4);

    // ---- dueling heads + mean(adv) + q
    fill_kernel<<<1, 1, 0, stream>>>(asum, 0.0f, 1);
    heads_kernel<<<blk(n, 8), 256, 0, stream>>>(bufA, Wv1, bv1, Wv2, bv2,
                                                Wa1, ba1, Wa2, ba2,
                                                val, adv, asum, n);
    q_kernel<<<blk(n, T), T, 0, stream>>>(val, adv, asum, (float*)d_out, n);
}